// TwoHeadCRF_15101105013365
// MI455X (gfx1250) — compile-verified
//
#include <hip/hip_runtime.h>
#include <stdint.h>

// Two-head masked Viterbi decode (CRF.decode) + one-hot expansion for MI455X.
// Max-plus recurrence cannot map to WMMA (no max-reduce matrix op); the CDNA5
// features used are the Tensor Data Mover (tensor_load_to_lds / TENSORcnt) for
// all chain-adjacent staging, wave32 shuffles, and LDS broadcast buffers.

#define BDIM 256
#define SDIM 4096
#define CDIM 16
#define CCAP 3
#define CH   128
#define NCH  (SDIM / CH)

typedef unsigned int       u32;
typedef unsigned long long u64;
typedef unsigned char      u8;

typedef unsigned int u32x4 __attribute__((ext_vector_type(4)));
typedef int          i32x8 __attribute__((ext_vector_type(8)));
typedef int          i32x4 __attribute__((ext_vector_type(4)));

// ---- Tensor Data Mover: 1-D contiguous load of `ndw` dwords into LDS. ----
// D# built per CDNA5 ISA ch.8: group0 {count=1, lds_addr, global_addr, type=2},
// group1 {data_size=4B, tensor_dim0=ndw, tensor_dim1=1, tile_dim0=ndw,
//         tile_dim1=1, tensor_dim0_stride=ndw}.  Groups 2/3 zero (<=2D tensor).
__device__ __forceinline__ void tdm_load_dw(void* lds_dst, const void* gsrc, u32 ndw) {
  u64 ga = (u64)(uintptr_t)gsrc;
  u32 lo = (u32)(uintptr_t)lds_dst;   // generic addr low 32 bits == LDS offset
  u32x4 g0 = { 1u,                                   // count=1, no gather
               lo,                                   // lds_addr
               (u32)ga,                              // global_addr[31:0]
               ((u32)(ga >> 32) & 0x01FFFFFFu) | (2u << 30) };  // addr[56:32]|type=2
  i32x8 g1 = { (int)(2u << 16),                      // data_size = 4 bytes
               (int)((ndw & 0xFFFFu) << 16),         // tensor_dim0[15:0]
               (int)((ndw >> 16) | (1u << 16)),      // tensor_dim0[31:16] | dim1 lo=1
               (int)((ndw & 0xFFFFu) << 16),         // dim1 hi=0 | tile_dim0
               1,                                    // tile_dim1=1, tile_dim2=0
               (int)ndw,                             // tensor_dim0_stride lo
               0, 0 };
  i32x4 z4 = { 0, 0, 0, 0 };
#if defined(__clang_major__) && (__clang_major__ >= 23)
  i32x8 z8 = { 0, 0, 0, 0, 0, 0, 0, 0 };
  __builtin_amdgcn_tensor_load_to_lds(g0, g1, z4, z4, z8, 0);
#else
  __builtin_amdgcn_tensor_load_to_lds(g0, g1, z4, z4, 0);
#endif
  __asm__ __volatile__("" ::: "memory");
}

// =====================  Forward (scores + backpointers)  =====================
// One block per batch element.  Wave0: 16-state head (lane=(half,i-range,j)).
// Wave1: 3-state head.  Emissions/mask TDM-double-buffered per 128-step chunk.
__global__ __launch_bounds__(64) void crf_forward(
    const float* __restrict__ x, const float* __restrict__ xcap,
    const int* __restrict__ amask,
    const float* __restrict__ startT, const float* __restrict__ trans,
    const float* __restrict__ endT,
    const float* __restrict__ startTc, const float* __restrict__ transc,
    const float* __restrict__ endTc,
    u64* __restrict__ bp_m, u8* __restrict__ bp_c,
    int* __restrict__ last_m, int* __restrict__ last_c)
{
  __shared__ __align__(16) float emis[2][CH * CDIM];   // 16 KB
  __shared__ __align__(16) float cemis[2][CH * CCAP];  // 3 KB
  __shared__ __align__(16) int   msk[2][CH];           // 1 KB
  __shared__ __align__(16) float sbuf[32];
  __shared__ __align__(16) float sbufc[32];
  __shared__ __align__(16) u8    bpb[CH][32];          // 4 KB
  __shared__ __align__(16) u8    cbpb[CH][32];         // 4 KB

  const int tid  = threadIdx.x;
  const int b    = blockIdx.x;
  const int lane = tid & 31;
  const bool w0  = tid < 32;

  int j = lane & 15, h = lane >> 4, ib = h * 8;
  float col0=0,col1=0,col2=0,col3=0,col4=0,col5=0,col6=0,col7=0, stv=0;
  int jc = 0; float cc0=0, cc1=0, cc2=0, cstv=0;
  if (w0) {
    col0 = trans[(ib+0)*CDIM + j]; col1 = trans[(ib+1)*CDIM + j];
    col2 = trans[(ib+2)*CDIM + j]; col3 = trans[(ib+3)*CDIM + j];
    col4 = trans[(ib+4)*CDIM + j]; col5 = trans[(ib+5)*CDIM + j];
    col6 = trans[(ib+6)*CDIM + j]; col7 = trans[(ib+7)*CDIM + j];
    stv  = startT[j];
  } else {
    jc  = lane % 3;
    cc0 = transc[0*CCAP + jc]; cc1 = transc[1*CCAP + jc]; cc2 = transc[2*CCAP + jc];
    cstv = startTc[jc];
  }

  // prologue: stage chunk 0
  if (w0) {
    tdm_load_dw(&emis[0][0], x + (size_t)b * SDIM * CDIM, CH * CDIM);
    tdm_load_dw(&msk[0][0],  amask + (size_t)b * SDIM,    CH);
  } else {
    tdm_load_dw(&cemis[0][0], xcap + (size_t)b * SDIM * CCAP, CH * CCAP);
  }
  __builtin_amdgcn_s_wait_tensorcnt(0);
  __syncthreads();

  float my = 0.0f, myc = 0.0f;

  for (int c = 0; c < NCH; ++c) {
    const int pb = c & 1, t0 = c * CH;
    if (c + 1 < NCH) {                    // async prefetch next chunk
      const int nb = (c + 1) & 1, t1 = (c + 1) * CH;
      if (w0) {
        tdm_load_dw(&emis[nb][0], x + ((size_t)b*SDIM + t1)*CDIM, CH*CDIM);
        tdm_load_dw(&msk[nb][0],  amask + (size_t)b*SDIM + t1,    CH);
      } else {
        tdm_load_dw(&cemis[nb][0], xcap + ((size_t)b*SDIM + t1)*CCAP, CH*CCAP);
      }
    }
    if (w0) {
      int tc = 0;
      if (c == 0) {                       // t = 0 init
        my = stv + emis[0][j];
        bpb[0][lane] = (u8)j;
        sbuf[lane] = my;
        __builtin_amdgcn_wave_barrier();
        tc = 1;
      }
      for (; tc < CH; ++tc) {
        float4 sa = *(const float4*)&sbuf[ib];
        float4 sb = *(const float4*)&sbuf[ib + 4];
        float e = emis[pb][tc*CDIM + j];
        int   m = msk[pb][tc];
        float v0 = sa.x + col0, v1 = sa.y + col1, v2 = sa.z + col2, v3 = sa.w + col3;
        float v4 = sb.x + col4, v5 = sb.y + col5, v6 = sb.z + col6, v7 = sb.w + col7;
        bool g;                            // first-index-wins argmax tree
        g = v0 >= v1;  float w01 = g?v0:v1;  int q01 = g?ib  :ib+1;
        g = v2 >= v3;  float w23 = g?v2:v3;  int q23 = g?ib+2:ib+3;
        g = v4 >= v5;  float w45 = g?v4:v5;  int q45 = g?ib+4:ib+5;
        g = v6 >= v7;  float w67 = g?v6:v7;  int q67 = g?ib+6:ib+7;
        g = w01 >= w23; float u03 = g?w01:w23; int p03 = g?q01:q23;
        g = w45 >= w67; float u47 = g?w45:w67; int p47 = g?q45:q67;
        g = u03 >= u47; float hv  = g?u03:u47; int hi  = g?p03:p47;
        float ov = __shfl_xor(hv, 16, 32);  // merge halves (half0 wins ties)
        int   oi = __shfl_xor(hi, 16, 32);
        float av = h ? ov : hv;  float bv = h ? hv : ov;
        int   ai = h ? oi : hi;  int   bi = h ? hi : oi;
        g = av >= bv; float red = g?av:bv; int ridx = g?ai:bi;
        float ns = red + e;
        bool live = (m != 0);               // masked step: carry + identity bp
        my = live ? ns : my;
        int bp = live ? ridx : j;
        bpb[tc][lane] = (u8)bp;
        sbuf[lane] = my;
        __builtin_amdgcn_wave_barrier();
      }
      // pack 16x4-bit backpointers per step (v_perm) and flush coalesced
      for (int r = 0; r < CH/32; ++r) {
        int row = r*32 + lane;
        uint4 d = *(const uint4*)&bpb[row][0];
        u32 p01 = d.x | (d.x >> 4);
        u32 p23 = d.y | (d.y >> 4);
        u32 p45 = d.z | (d.z >> 4);
        u32 p67 = d.w | (d.w >> 4);
        u32 lo  = __builtin_amdgcn_perm(p23, p01, 0x06040200u);
        u32 hi2 = __builtin_amdgcn_perm(p67, p45, 0x06040200u);
        bp_m[(size_t)b*SDIM + t0 + row] = ((u64)hi2 << 32) | lo;
      }
    } else {                               // ---- cap head, 3 states ----
      int tc = 0;
      if (c == 0) {
        myc = cstv + cemis[0][jc];
        cbpb[0][lane] = (u8)jc;
        sbufc[lane] = myc;
        __builtin_amdgcn_wave_barrier();
        tc = 1;
      }
      for (; tc < CH; ++tc) {
        float s0 = sbufc[0], s1 = sbufc[1], s2 = sbufc[2];
        float e = cemis[pb][tc*CCAP + jc];
        int   m = msk[pb][tc];
        float v0 = s0 + cc0, v1 = s1 + cc1, v2 = s2 + cc2;
        float bv = v0; int bi = 0;
        bool g = v1 > bv; bv = g?v1:bv; bi = g?1:bi;
        g = v2 > bv;      bv = g?v2:bv; bi = g?2:bi;
        float ns = bv + e;
        bool live = (m != 0);
        myc = live ? ns : myc;
        int bp = live ? bi : jc;
        cbpb[tc][lane] = (u8)bp;
        sbufc[lane] = myc;
        __builtin_amdgcn_wave_barrier();
      }
      for (int r = 0; r < CH/32; ++r) {     // pack 3x2-bit, flush
        int row = r*32 + lane;
        u32 d = *(const u32*)&cbpb[row][0];
        u32 pk = (d & 3u) | (((d >> 8) & 3u) << 2) | (((d >> 16) & 3u) << 4);
        bp_c[(size_t)b*SDIM + t0 + row] = (u8)pk;
      }
    }
    __builtin_amdgcn_s_wait_tensorcnt(0);
    __syncthreads();
  }

  // argmax(final_score + end_transitions), first-max semantics
  if (tid == 0) {
    float best = sbuf[0] + endT[0]; int bi = 0;
    for (int i = 1; i < CDIM; ++i) {
      float f = sbuf[i] + endT[i];
      if (f > best) { best = f; bi = i; }
    }
    last_m[b] = bi;
  }
  if (tid == 32) {
    float best = sbufc[0] + endTc[0]; int bi = 0;
    for (int i = 1; i < CCAP; ++i) {
      float f = sbufc[i] + endTc[i];
      if (f > best) { best = f; bi = i; }
    }
    last_c[b] = bi;
  }
}

// ==================  Backtrack + one-hot output expansion  ==================
// One wave per batch.  bp words TDM-prefetched; the serial chain is pure VALU
// (select dword, shift, mask).  Outputs flushed cooperatively, coalesced b128.
__global__ __launch_bounds__(32) void crf_backtrack(
    const u64* __restrict__ bp_m, const u8* __restrict__ bp_c,
    const int* __restrict__ amask,
    const int* __restrict__ last_m, const int* __restrict__ last_c,
    float* __restrict__ y, float* __restrict__ ycap)
{
  __shared__ __align__(16) u64 wbuf[2][CH];            // 2 KB
  __shared__ __align__(16) u8  cbuf[2][CH];            // 256 B
  __shared__ __align__(16) int mbuf[2][CH];            // 1 KB
  __shared__ __align__(16) unsigned short tagbuf[CH][32];  // 8 KB

  const int b = blockIdx.x, lane = threadIdx.x;
  int tag  = last_m[b];
  int tagc = last_c[b];

  {   // prologue: stage last chunk
    const int cc = NCH - 1, t0 = cc * CH, pb = cc & 1;
    tdm_load_dw(&wbuf[pb][0], bp_m + (size_t)b*SDIM + t0, CH * 2);
    tdm_load_dw(&cbuf[pb][0], bp_c + (size_t)b*SDIM + t0, CH / 4);
    tdm_load_dw(&mbuf[pb][0], amask + (size_t)b*SDIM + t0, CH);
    __builtin_amdgcn_s_wait_tensorcnt(0);
    __builtin_amdgcn_wave_barrier();
  }

  for (int c = NCH - 1; c >= 0; --c) {
    const int pb = c & 1, t0 = c * CH;
    if (c > 0) {                          // async prefetch previous chunk
      const int nb = (c - 1) & 1, t1 = (c - 1) * CH;
      tdm_load_dw(&wbuf[nb][0], bp_m + (size_t)b*SDIM + t1, CH * 2);
      tdm_load_dw(&cbuf[nb][0], bp_c + (size_t)b*SDIM + t1, CH / 4);
      tdm_load_dw(&mbuf[nb][0], amask + (size_t)b*SDIM + t1, CH);
    }
    for (int tc = CH - 1; tc >= 0; --tc) {  // serial pointer chase (replicated)
      const int t = t0 + tc;
      int m = mbuf[pb][tc];
      int tmo = m ? tag  : 0;               // where(mask, tags, 0)
      int tco = m ? tagc : 0;
      tagbuf[tc][lane] = (unsigned short)(tmo | (tco << 8));
      if (t > 0) {
        u64 w  = wbuf[pb][tc];
        u32 hw = (tag & 8) ? (u32)(w >> 32) : (u32)w;
        tag  = (int)((hw >> ((tag & 7) * 4)) & 15u);
        u32 cw = cbuf[pb][tc];
        tagc = (int)((cw >> (2 * tagc)) & 3u);
      }
    }
    __builtin_amdgcn_wave_barrier();
    for (int r = 0; r < CH/32; ++r) {       // cooperative one-hot flush
      int row = r*32 + lane;
      int t = t0 + row;
      u32 v = tagbuf[row][0];
      int tm = (int)(v & 255u), tv = (int)(v >> 8);
      size_t ob = ((size_t)b*SDIM + t) * CDIM;
      float4 o;
      o.x=(float)(tm==0);  o.y=(float)(tm==1);  o.z=(float)(tm==2);  o.w=(float)(tm==3);
      *(float4*)&y[ob + 0] = o;
      o.x=(float)(tm==4);  o.y=(float)(tm==5);  o.z=(float)(tm==6);  o.w=(float)(tm==7);
      *(float4*)&y[ob + 4] = o;
      o.x=(float)(tm==8);  o.y=(float)(tm==9);  o.z=(float)(tm==10); o.w=(float)(tm==11);
      *(float4*)&y[ob + 8] = o;
      o.x=(float)(tm==12); o.y=(float)(tm==13); o.z=(float)(tm==14); o.w=(float)(tm==15);
      *(float4*)&y[ob + 12] = o;
      size_t cb = ((size_t)b*SDIM + t) * CCAP;
      ycap[cb + 0] = (float)(tv==0);
      ycap[cb + 1] = (float)(tv==1);
      ycap[cb + 2] = (float)(tv==2);
    }
    __builtin_amdgcn_s_wait_tensorcnt(0);
    __builtin_amdgcn_wave_barrier();
  }
}

extern "C" void kernel_launch(void* const* d_in, const int* in_sizes, int n_in,
                              void* d_out, int out_size, void* d_ws, size_t ws_size,
                              hipStream_t stream) {
  (void)in_sizes; (void)n_in; (void)out_size; (void)ws_size;
  const float* x      = (const float*)d_in[0];
  const float* xcap   = (const float*)d_in[1];
  const int*   amask  = (const int*)  d_in[2];
  const float* startT = (const float*)d_in[3];
  const float* trans  = (const float*)d_in[4];
  const float* endT   = (const float*)d_in[5];
  const float* startTc= (const float*)d_in[6];
  const float* transc = (const float*)d_in[7];
  const float* endTc  = (const float*)d_in[8];

  char* ws = (char*)d_ws;                       // ~9.5 MB scratch used
  u64* bpm  = (u64*)ws;                                   // [B*S] packed 16x4b
  u8*  bpc  = (u8*)(ws + (size_t)BDIM*SDIM*8);            // [B*S] packed 3x2b
  int* lastm = (int*)(ws + (size_t)BDIM*SDIM*9);          // [B]
  int* lastc = (int*)(ws + (size_t)BDIM*SDIM*9 + 4096);   // [B]

  float* y    = (float*)d_out;                            // [B,S,16]
  float* ycap = y + (size_t)BDIM*SDIM*CDIM;               // [B,S,3]

  crf_forward<<<dim3(BDIM), dim3(64), 0, stream>>>(
      x, xcap, amask, startT, trans, endT, startTc, transc, endTc,
      bpm, bpc, lastm, lastc);
  crf_backtrack<<<dim3(BDIM), dim3(32), 0, stream>>>(
      bpm, bpc, amask, lastm, lastc, y, ycap);
}